// SE3TransformerTrIP_67989332295700
// MI455X (gfx1250) — compile-verified
//
#include <hip/hip_runtime.h>
#include <math.h>

// ---------------------------------------------------------------------------
// SE(3)-Transformer (TrIP) forward for MI455X / gfx1250.
//
// Heavy math = per-edge radial matmuls R = relu(r*W1+b1) @ W2^T, mapped onto
// chained V_WMMA_F32_16X16X4_F32 (M=16 edges, K=32 hidden, N=16-col tiles of
// the 512/1536-wide output).  ~252 GFLOP total, compute-bound (traffic is only
// ~0.7 GB vs 23.3 TB/s HBM; weights/feats are L2-resident).  f32 WMMA keeps
// exact reference semantics through 2 attention layers.
//
// Input-order assumption (setup_inputs dict order, params flattened as a JAX
// pytree with sorted dict keys):
//   0 src, 1 dst, 2 rel_pos, 3 scale, 4 node0, 5 node1,
//   6..13  final: 00.{W1,W2,b1,b2}, 10.{W1,W2,b1,b2}
//   14+36L per layer: Wp0,Wp1,Wq0,Wq1, key.{00,01,10,11}.{W1,W2,b1,b2},
//                     value.{00,01,10,11}.{W1,W2,b1,b2}
// ---------------------------------------------------------------------------

typedef float v2f __attribute__((ext_vector_type(2)));
typedef float v8f __attribute__((ext_vector_type(8)));

#define CH_ 32
#define CV_ 16
#define Y0C 0.28209479177387814f
#define C1C 0.4886025119029199f
#define INV_SQRT8 0.35355339059327373f   // 1/sqrt(CHH*4), CHH=2

struct MLPp  { const float* W1; const float* b1; const float* W2; const float* b2; };
struct LayerP { MLPp mlp[2][4]; };        // [0]=value, [1]=key ; pairs 00,01,10,11
struct FinalP { MLPp m00; MLPp m10; };

__device__ __forceinline__ void atomicMaxFloat(float* addr, float val) {
  if (val >= 0.0f) atomicMax((int*)addr, __float_as_int(val));
  else             atomicMin((unsigned int*)addr, __float_as_uint(val));
}

// ---------------------------------------------------------------- init -----
__global__ void init_attn_kernel(float* zmax, float* denom, float* agg0,
                                 float* agg1, int N) {
  int t = blockIdx.x * blockDim.x + threadIdx.x;
  if (t < N * 8)  { zmax[t] = -INFINITY; return; }  t -= N * 8;
  if (t < N * 8)  { denom[t] = 0.0f;     return; }  t -= N * 8;
  if (t < N * 16) { agg0[t]  = 0.0f;     return; }  t -= N * 16;
  if (t < N * 48) { agg1[t]  = 0.0f; }
}

// ------------------------------------------------------- q projection -----
__global__ void qproj_kernel(const float* __restrict__ f0, const float* __restrict__ f1,
                             const float* __restrict__ Wq0, const float* __restrict__ Wq1,
                             float* __restrict__ q0, float* __restrict__ q1, int N) {
  int tid = blockIdx.x * blockDim.x + threadIdx.x;
  if (tid >= N * 16) return;
  int n = tid >> 4, o = tid & 15;
  float a0 = 0.0f, a1[3] = {0.0f, 0.0f, 0.0f};
  for (int c = 0; c < 32; ++c) {
    float w0 = Wq0[o * 32 + c], w1 = Wq1[o * 32 + c];
    a0 += w0 * f0[n * 32 + c];
#pragma unroll
    for (int m = 0; m < 3; ++m) a1[m] += w1 * f1[n * 96 + c * 3 + m];
  }
  q0[n * 16 + o] = a0;
#pragma unroll
  for (int m = 0; m < 3; ++m) q1[n * 48 + o * 3 + m] = a1[m];
}

// ------------------------------------------------ edge k/v WMMA kernel -----
// One wave32 per 16-edge block.  EXEC is all-ones at every WMMA.
__launch_bounds__(32)
__global__ void edge_kv_kernel(const int* __restrict__ src, const int* __restrict__ dst,
                               const float* __restrict__ rel_pos, const float* __restrict__ scale,
                               const float* __restrict__ f0, const float* __restrict__ f1,
                               LayerP P,
                               const float* __restrict__ q0, const float* __restrict__ q1,
                               float* __restrict__ zmax, float* __restrict__ zbuf,
                               float* __restrict__ v0o, float* __restrict__ v1o, int E) {
  __shared__ float sB11[16][27];      // basis '11' [n][m][j]
  __shared__ float sb1v[16][3];       // C1*u*scale
  __shared__ float sb00[16];
  __shared__ float sr[16];
  __shared__ int   ssrc[16], sdst[16];
  __shared__ float sf0[16][32];
  __shared__ float sf1[16][32][3];
  __shared__ float st[16][32][9];     // t[c][n*nj+j], reused per pair
  __shared__ float sh[16][32];        // radial hidden H (WMMA A source)
  __shared__ __align__(32) float sR[32][16];  // two 16x16 R tiles (C-layout)
  __shared__ float sv0[16][16], sk0[16][16];
  __shared__ float sv1[16][16][3], sk1[16][16][3];

  const int lane  = threadIdx.x;
  const int eBase = blockIdx.x * 16;

  if (lane < 16) {
    int e = eBase + lane; if (e >= E) e = E - 1;
    float x = rel_pos[e * 3 + 0], y = rel_pos[e * 3 + 1], z = rel_pos[e * 3 + 2];
    float rr  = sqrtf(x * x + y * y + z * z);
    float inv = 1.0f / (rr + 1e-9f);
    // u permuted [1,2,0]
    float u0 = y * inv, u1 = z * inv, u2 = x * inv;
    float sc = scale[e];
    sr[lane] = rr;
    sb00[lane] = Y0C * sc;
    sb1v[lane][0] = C1C * u0 * sc;
    sb1v[lane][1] = C1C * u1 * sc;
    sb1v[lane][2] = C1C * u2 * sc;
    float u[3] = {u0, u1, u2};
#pragma unroll
    for (int n = 0; n < 3; ++n)
#pragma unroll
      for (int m = 0; m < 3; ++m) {
        float j0 = (n == m) ? Y0C : 0.0f;
        float j2 = u[n] * u[m] - ((n == m) ? (1.0f / 3.0f) : 0.0f);
        sB11[lane][(n * 3 + m) * 3 + 0] = j0 * sc;
        sB11[lane][(n * 3 + m) * 3 + 2] = j2 * sc;
      }
    // j=1: eps contraction  B[a][b] = sum_c eps[a][b][c] u[c]
    sB11[lane][(0*3+0)*3+1] = 0.0f;        sB11[lane][(1*3+1)*3+1] = 0.0f;
    sB11[lane][(2*3+2)*3+1] = 0.0f;
    sB11[lane][(0*3+1)*3+1] =  u2 * sc;    sB11[lane][(0*3+2)*3+1] = -u1 * sc;
    sB11[lane][(1*3+0)*3+1] = -u2 * sc;    sB11[lane][(1*3+2)*3+1] =  u0 * sc;
    sB11[lane][(2*3+0)*3+1] =  u1 * sc;    sB11[lane][(2*3+1)*3+1] = -u0 * sc;
    ssrc[lane] = src[e];
    sdst[lane] = dst[e];
  }
  __syncthreads();

  for (int i = lane; i < 16 * 32; i += 32) {
    int e = i >> 5, c = i & 31;
    sf0[e][c] = f0[ssrc[e] * 32 + c];
  }
  for (int i = lane; i < 16 * 96; i += 32) {
    int e = i / 96, cm = i % 96;
    sf1[e][cm / 3][cm % 3] = f1[ssrc[e] * 96 + cm];
  }
  for (int i = lane; i < 16 * 16; i += 32) {
    int e = i >> 4, o = i & 15;
    sv0[e][o] = 0.0f; sk0[e][o] = 0.0f;
#pragma unroll
    for (int m = 0; m < 3; ++m) { sv1[e][o][m] = 0.0f; sk1[e][o][m] = 0.0f; }
  }
  __syncthreads();

#pragma unroll
  for (int p = 0; p < 4; ++p) {                 // pairs 00,01,10,11
    const int nj   = (p == 3) ? 3 : 1;          // basis j-dim
    const int nn   = (p == 0 || p == 2) ? 1 : 3;// output degree dim
    const int dego = (p == 1 || p == 3);

    // ---- build t[c][n][j] = sum_m B[n][m][j] * f[c][m]
    for (int i = lane; i < 16 * 32; i += 32) {
      int e = i >> 5, c = i & 31;
      if (p == 0) {
        st[e][c][0] = sb00[e] * sf0[e][c];
      } else if (p == 1) {
#pragma unroll
        for (int n = 0; n < 3; ++n) st[e][c][n] = sb1v[e][n] * sf0[e][c];
      } else if (p == 2) {
        float s = 0.0f;
#pragma unroll
        for (int m = 0; m < 3; ++m) s += sb1v[e][m] * sf1[e][c][m];
        st[e][c][0] = s;
      } else {
#pragma unroll
        for (int n = 0; n < 3; ++n)
#pragma unroll
          for (int j = 0; j < 3; ++j) {
            float s = 0.0f;
#pragma unroll
            for (int m = 0; m < 3; ++m) s += sB11[e][(n * 3 + m) * 3 + j] * sf1[e][c][m];
            st[e][c][n * 3 + j] = s;
          }
      }
    }
    __syncthreads();

    for (int vk = 0; vk < 2; ++vk) {            // 0=value, 1=key
      MLPp mp = P.mlp[vk][p];
      // radial hidden: h[e][k] = relu(r*W1[k] + b1[k])
      for (int i = lane; i < 16 * 32; i += 32) {
        int e = i >> 5, k = i & 31;
        sh[e][k] = fmaxf(sr[e] * mp.W1[k] + mp.b1[k], 0.0f);
      }
      __syncthreads();

      // A operand (16 edges x 32 k), preloaded once, reused across all tiles.
      // f32 16x16x4 A layout: lane<16 -> K{0,1}, lane>=16 -> K{2,3}; M=lane%16
      const int Me = lane & 15, hi2 = (lane >> 4) * 2;
      v2f A[8];
#pragma unroll
      for (int ks = 0; ks < 8; ++ks) {
        v2f a; a.x = sh[Me][ks * 4 + hi2]; a.y = sh[Me][ks * 4 + hi2 + 1];
        A[ks] = a;
      }

      const int ntiles = 32 * nj;               // ODIM/16 = 512*nj/16
      const int tpo    = 2 * nj;                // 16-col tiles per output ch o
      for (int Tb = 0; Tb < ntiles; Tb += 2) {  // 2 independent WMMA chains
        v8f acc0 = {}; v8f acc1 = {};
        const float* w2p0 = mp.W2 + (Tb * 16 + Me) * 32 + hi2;
        const float* w2p1 = w2p0 + 16 * 32;
#pragma unroll
        for (int ks = 0; ks < 8; ++ks) {
          v2f b0 = *(const v2f*)(w2p0 + ks * 4);
          v2f b1 = *(const v2f*)(w2p1 + ks * 4);
          acc0 = __builtin_amdgcn_wmma_f32_16x16x4_f32(false, A[ks], false, b0,
                                                       (short)0, acc0, false, false);
          acc1 = __builtin_amdgcn_wmma_f32_16x16x4_f32(false, A[ks], false, b1,
                                                       (short)0, acc1, false, false);
        }
        *(v8f*)(&sR[lane][0]) = acc0;           // C layout: vgpr r, lane<16: M=r
        *(v8f*)(&sR[lane][8]) = acc1;           //           lane>=16: M=r+8
        __syncthreads();

        if (lane < 16) {
          int e = lane;
#pragma unroll
          for (int tt = 0; tt < 2; ++tt) {
            int T   = Tb + tt;
            int o   = T / tpo;
            int cjb = (T % tpo) * 16;
            const float* b2p = mp.b2 + T * 16;
            float a0 = 0.0f, a1 = 0.0f, a2 = 0.0f;
#pragma unroll
            for (int col = 0; col < 16; ++col) {
              int cj = cjb + col;
              int c  = cj / nj, j = cj - c * nj;
              float Rv = sR[col + ((e & 8) ? 16 : 0)][tt * 8 + (e & 7)] + b2p[col];
              const float* tp = &st[e][c][0];
              a0 += Rv * tp[j];
              if (nn == 3) { a1 += Rv * tp[nj + j]; a2 += Rv * tp[2 * nj + j]; }
            }
            if (dego == 0) {
              float* op = (vk == 0) ? &sv0[e][o] : &sk0[e][o];
              op[0] += a0;
            } else {
              float* op = (vk == 0) ? &sv1[e][o][0] : &sk1[e][o][0];
              op[0] += a0; op[1] += a1; op[2] += a2;
            }
          }
        }
        __syncthreads();
      }
    }
    __syncthreads();
  }

  // ---- write v, compute attention logits z with q[dst], seed segment-max
  for (int i = lane; i < 16 * 16; i += 32) {
    int e = i >> 4, o = i & 15;
    int ed = eBase + e;
    if (ed < E) {
      v0o[ed * 16 + o] = sv0[e][o];
#pragma unroll
      for (int m = 0; m < 3; ++m) v1o[ed * 48 + o * 3 + m] = sv1[e][o][m];
    }
  }
  for (int i = lane; i < 16 * 8; i += 32) {
    int e = i >> 3, h = i & 7;
    int ed = eBase + e;
    if (ed < E) {
      int d = sdst[e];
      float z = 0.0f;
#pragma unroll
      for (int c = 0; c < 2; ++c) {
        int o = h * 2 + c;
        z += sk0[e][o] * q0[d * 16 + o];
#pragma unroll
        for (int m = 0; m < 3; ++m) z += sk1[e][o][m] * q1[d * 48 + o * 3 + m];
      }
      z *= INV_SQRT8;
      zbuf[ed * 8 + h] = z;
      atomicMaxFloat(&zmax[d * 8 + h], z);
    }
  }
}

// -------------------------------------------------- softmax numerator -----
__global__ void softmax_ez_kernel(const int* __restrict__ dst, const float* __restrict__ scale,
                                  const float* __restrict__ zmax, float* __restrict__ zbuf,
                                  float* __restrict__ denom, int E) {
  int tid = blockIdx.x * blockDim.x + threadIdx.x;
  if (tid >= E * 8) return;
  int e = tid >> 3, h = tid & 7;
  int d = dst[e];
  float ez = scale[e] * expf(zbuf[tid] - zmax[d * 8 + h]);
  zbuf[tid] = ez;                      // overwrite z with ez
  atomicAdd(&denom[d * 8 + h], ez);
}

// -------------------------------------------------------- aggregation -----
__global__ void aggregate_kernel(const int* __restrict__ dst, const float* __restrict__ ez,
                                 const float* __restrict__ denom,
                                 const float* __restrict__ v0, const float* __restrict__ v1,
                                 float* __restrict__ agg0, float* __restrict__ agg1, int E) {
  int tid = blockIdx.x * blockDim.x + threadIdx.x;
  if (tid >= E * 16) return;
  int e = tid >> 4, o = tid & 15, h = o >> 1;
  int d = dst[e];
  float alpha = ez[e * 8 + h] / (denom[d * 8 + h] + 1e-9f);
  atomicAdd(&agg0[d * 16 + o], alpha * v0[e * 16 + o]);
#pragma unroll
  for (int m = 0; m < 3; ++m)
    atomicAdd(&agg1[d * 48 + o * 3 + m], alpha * v1[e * 48 + o * 3 + m]);
}

// -------------------------------------------------- output projection -----
__global__ void proj_kernel(const float* __restrict__ agg0, const float* __restrict__ agg1,
                            const float* __restrict__ f0, const float* __restrict__ f1,
                            const float* __restrict__ Wp0, const float* __restrict__ Wp1,
                            float* __restrict__ nf0, float* __restrict__ nf1, int N) {
  int tid = blockIdx.x * blockDim.x + threadIdx.x;
  if (tid >= N * 32) return;
  int n = tid >> 5, o = tid & 31;
  float a0 = 0.0f, a1[3] = {0.0f, 0.0f, 0.0f};
  for (int c = 0; c < 16; ++c) {       // cat = [agg(16), feats(32)]
    float w0 = Wp0[o * 48 + c], w1 = Wp1[o * 48 + c];
    a0 += w0 * agg0[n * 16 + c];
#pragma unroll
    for (int m = 0; m < 3; ++m) a1[m] += w1 * agg1[n * 48 + c * 3 + m];
  }
  for (int c = 0; c < 32; ++c) {
    float w0 = Wp0[o * 48 + 16 + c], w1 = Wp1[o * 48 + 16 + c];
    a0 += w0 * f0[n * 32 + c];
#pragma unroll
    for (int m = 0; m < 3; ++m) a1[m] += w1 * f1[n * 96 + c * 3 + m];
  }
  nf0[n * 32 + o] = a0;
#pragma unroll
  for (int m = 0; m < 3; ++m) nf1[n * 96 + o * 3 + m] = a1[m];
}

// --------------------------------------------------------- final head -----
__global__ void final_kernel(const int* __restrict__ src, const float* __restrict__ rel_pos,
                             const float* __restrict__ scale,
                             const float* __restrict__ f0, const float* __restrict__ f1,
                             FinalP FP, float* __restrict__ out, int E) {
  int e = blockIdx.x * blockDim.x + threadIdx.x;
  if (e >= E) return;
  float x = rel_pos[e * 3 + 0], y = rel_pos[e * 3 + 1], z = rel_pos[e * 3 + 2];
  float rr  = sqrtf(x * x + y * y + z * z);
  float inv = 1.0f / (rr + 1e-9f);
  float u0 = y * inv, u1 = z * inv, u2 = x * inv;   // permuted [1,2,0]
  float sc = scale[e];
  int s = src[e];
  float h[32];
  // pair '00'
  for (int k = 0; k < 32; ++k) h[k] = fmaxf(rr * FP.m00.W1[k] + FP.m00.b1[k], 0.0f);
  float b00 = Y0C * sc;
  float acc = 0.0f;
  for (int c = 0; c < 32; ++c) {
    float R = FP.m00.b2[c];
    for (int k = 0; k < 32; ++k) R += FP.m00.W2[c * 32 + k] * h[k];
    acc += R * (b00 * f0[s * 32 + c]);
  }
  // pair '10'
  for (int k = 0; k < 32; ++k) h[k] = fmaxf(rr * FP.m10.W1[k] + FP.m10.b1[k], 0.0f);
  float bv0 = C1C * u0 * sc, bv1 = C1C * u1 * sc, bv2 = C1C * u2 * sc;
  for (int c = 0; c < 32; ++c) {
    float R = FP.m10.b2[c];
    for (int k = 0; k < 32; ++k) R += FP.m10.W2[c * 32 + k] * h[k];
    float t = bv0 * f1[s * 96 + c * 3 + 0] + bv1 * f1[s * 96 + c * 3 + 1]
            + bv2 * f1[s * 96 + c * 3 + 2];
    acc += R * t;
  }
  out[e] = acc;
}

// ------------------------------------------------------------- launch -----
extern "C" void kernel_launch(void* const* d_in, const int* in_sizes, int n_in,
                              void* d_out, int out_size, void* d_ws, size_t ws_size,
                              hipStream_t stream) {
  const int*   src     = (const int*)d_in[0];
  const int*   dst     = (const int*)d_in[1];
  const float* rel_pos = (const float*)d_in[2];
  const float* scale   = (const float*)d_in[3];
  const float* node0   = (const float*)d_in[4];
  const float* node1   = (const float*)d_in[5];
  const int E = in_sizes[0];
  const int N = in_sizes[4] / CH_;

  auto F = [&](int i) { return (const float*)d_in[i]; };

  FinalP FP;
  FP.m00 = { F(6),  F(8),  F(7),  F(9)  };   // W1, b1, W2, b2
  FP.m10 = { F(10), F(12), F(11), F(13) };

  LayerP LP[2];
  const float *Wp0[2], *Wp1[2], *Wq0[2], *Wq1[2];
  for (int L = 0; L < 2; ++L) {
    int b = 14 + 36 * L;
    Wp0[L] = F(b + 0); Wp1[L] = F(b + 1); Wq0[L] = F(b + 2); Wq1[L] = F(b + 3);
    for (int p = 0; p < 4; ++p) {
      int kb = b + 4  + 4 * p;   // key mlps   (sorted: 'key' < 'value')
      int vb = b + 20 + 4 * p;   // value mlps
      LP[L].mlp[1][p] = { F(kb + 0), F(kb + 2), F(kb + 1), F(kb + 3) };
      LP[L].mlp[0][p] = { F(vb + 0), F(vb + 2), F(vb + 1), F(vb + 3) };
    }
  }

  // workspace carve (floats); total ~108 MB
  float* w = (float*)d_ws;
  size_t off = 0;
  auto alloc = [&](size_t n) { float* p = w + off; off += n; return p; };
  float* q0    = alloc((size_t)N * 16);
  float* q1    = alloc((size_t)N * 48);
  float* zmax  = alloc((size_t)N * 8);
  float* denom = alloc((size_t)N * 8);
  float* agg0  = alloc((size_t)N * 16);
  float* agg1  = alloc((size_t)N * 48);
  float* zbuf  = alloc((size_t)E * 8);
  float* v0    = alloc((size_t)E * 16);
  float* v1    = alloc((size_t)E * 48);
  float* fA0   = alloc((size_t)N * 32);
  float* fA1   = alloc((size_t)N * 96);
  float* fB0   = alloc((size_t)N * 32);
  float* fB1   = alloc((size_t)N * 96);

  const float* cf0 = node0;
  const float* cf1 = node1;
  float* buf0[2] = { fA0, fB0 };
  float* buf1[2] = { fA1, fB1 };

  for (int L = 0; L < 2; ++L) {
    init_attn_kernel<<<(N * 80 + 255) / 256, 256, 0, stream>>>(zmax, denom, agg0, agg1, N);
    qproj_kernel<<<(N * 16 + 255) / 256, 256, 0, stream>>>(cf0, cf1, Wq0[L], Wq1[L], q0, q1, N);
    edge_kv_kernel<<<(E + 15) / 16, 32, 0, stream>>>(src, dst, rel_pos, scale, cf0, cf1,
                                                     LP[L], q0, q1, zmax, zbuf, v0, v1, E);
    softmax_ez_kernel<<<(E * 8 + 255) / 256, 256, 0, stream>>>(dst, scale, zmax, zbuf, denom, E);
    aggregate_kernel<<<(E * 16 + 255) / 256, 256, 0, stream>>>(dst, zbuf, denom, v0, v1,
                                                               agg0, agg1, E);
    proj_kernel<<<(N * 32 + 255) / 256, 256, 0, stream>>>(agg0, agg1, cf0, cf1,
                                                          Wp0[L], Wp1[L], buf0[L], buf1[L], N);
    cf0 = buf0[L];
    cf1 = buf1[L];
  }
  final_kernel<<<(E + 255) / 256, 256, 0, stream>>>(src, rel_pos, scale, cf0, cf1, FP,
                                                    (float*)d_out, E);
}